// ParallelMetaLA_Attention_selfaug_41747082117842
// MI455X (gfx1250) — compile-verified
//
#include <hip/hip_runtime.h>
#include <hip/hip_bf16.h>

// ---------------------------------------------------------------------------
// MetaLA / GLA forward for MI455X (gfx1250, wave32, WMMA).
// - All GEMMs use V_WMMA_F32_16X16X4_F32 (workload is HBM-bound at 23.3 TB/s,
//   so fp32 matrix path keeps reference numerics at no roofline cost).
// - The L=8192 recurrence runs in chunks of 64 steps -> pure WMMA GEMMs with
//   a 32x64 fp32 state carried in LDS between chunks (128 independent (b,h)
//   workgroups).
// - All tile staging uses CDNA5 async global->LDS B128 copies (ASYNCcnt),
//   bypassing the VGPR file; data readiness via s_wait_asynccnt + barrier.
// ---------------------------------------------------------------------------

typedef float v2f __attribute__((ext_vector_type(2)));
typedef float v8f __attribute__((ext_vector_type(8)));

#define Lseq   8192
#define Bsz    8
#define Dmod   1024
#define Hh     16
#define DK     32
#define DV     64
#define CHUNK  64
#define NCHUNK (Lseq / CHUNK)
#define Mrows  (Bsz * Lseq)          // 65536 GEMM rows

__device__ __forceinline__ v8f wmma4(v2f a, v2f b, v8f c) {
  // (neg_a, A, neg_b, B, c_mod, C, reuse_a, reuse_b)
  return __builtin_amdgcn_wmma_f32_16x16x4_f32(false, a, false, b, (short)0, c,
                                               false, false);
}

// --- CDNA5 async global->LDS copy (VGLOBAL encoding, ASYNCcnt) -------------
// the LDS address comes from VGPR vdst; generic shared pointers carry the LDS byte
// offset in their low 32 bits.
__device__ __forceinline__ void async_b128(void* lds, const void* g) {
  uint32_t lo = (uint32_t)(uintptr_t)lds;
  uint64_t ga = (uint64_t)(uintptr_t)g;
  asm volatile("global_load_async_to_lds_b128 %0, %1, off"
               :: "v"(lo), "v"(ga) : "memory");
}
__device__ __forceinline__ void async_wait0() {
  asm volatile("s_wait_asynccnt 0x0" ::: "memory");
}

// --- WMMA fragment loaders for 16x16x4 f32, wave32 -------------------------
// A (16x4, MxK): lanes 0-15 hold K={k0,k0+1}, lanes 16-31 hold K={k0+2,k0+3};
// row M = lane&15. B (4x16, KxN): lanes 0-15 = rows k0/k0+1 in vgpr0/1,
// lanes 16-31 = rows k0+2/k0+3; col N = lane&15. C/D: vgpr i -> row i (lanes
// 0-15) or 8+i (lanes 16-31), col = lane&15.
__device__ __forceinline__ v2f ld_a(const float* s, int ld, int m0, int k0, int lane) {
  int m = lane & 15, kh = lane >> 4;
  const float* p = s + (m0 + m) * ld + k0 + kh * 2;
  v2f a; a.x = p[0]; a.y = p[1]; return a;
}
__device__ __forceinline__ v2f ld_b(const float* s, int ld, int k0, int n0, int lane) {
  int n = lane & 15, kh = lane >> 4;
  v2f b;
  b.x = s[(k0 + kh * 2 + 0) * ld + n0 + n];
  b.y = s[(k0 + kh * 2 + 1) * ld + n0 + n];
  return b;
}
// B[k][n] = s[n][k]   (transposed source)
__device__ __forceinline__ v2f ld_bt(const float* s, int ld, int k0, int n0, int lane) {
  int n = lane & 15, kh = lane >> 4;
  const float* p = s + (n0 + n) * ld + k0 + kh * 2;
  v2f b; b.x = p[0]; b.y = p[1]; return b;
}
// A[m][k] = s[k][m]   (transposed source)
__device__ __forceinline__ v2f ld_at(const float* s, int ld, int m0, int k0, int lane) {
  int m = lane & 15, kh = lane >> 4;
  v2f a;
  a.x = s[(k0 + kh * 2 + 0) * ld + m0 + m];
  a.y = s[(k0 + kh * 2 + 1) * ld + m0 + m];
  return a;
}

__device__ __forceinline__ float sigmoidf_(float x) { return 1.0f / (1.0f + expf(-x)); }
__device__ __forceinline__ float siluf_(float x)    { return x * sigmoidf_(x); }
__device__ __forceinline__ float logsigf_(float x) {
  float l = log1pf(expf(-fabsf(x)));
  return (x >= 0.0f) ? -l : (x - l);
}

// ---------------------------------------------------------------------------
// K1: causal depthwise conv (W=4) + SiLU ; x[L,B,D] -> xb[B,L,D]
// ---------------------------------------------------------------------------
__global__ __launch_bounds__(256) void k_conv_silu(const float* __restrict__ x,
                                                   const float* __restrict__ w,
                                                   float* __restrict__ xb) {
  size_t idx = (size_t)blockIdx.x * 256 + threadIdx.x;   // over B*L*D
  int d = (int)(idx & (Dmod - 1));
  size_t bl = idx >> 10;
  int l = (int)(bl & (Lseq - 1));
  int b = (int)(bl >> 13);
  float acc = 0.0f;
#pragma unroll
  for (int kk = 0; kk < 4; kk++) {
    int ls = l - 3 + kk;
    float xv = (ls >= 0) ? x[((size_t)ls * Bsz + b) * Dmod + d] : 0.0f;
    acc = fmaf(xv, w[d * 4 + kk], acc);
  }
  xb[idx] = siluf_(acc);
}

// ---------------------------------------------------------------------------
// K2: projection GEMM  C[65536,N] = xb[65536,1024] @ W[1024,N], WMMA f32.
// variant 0: q  -> [B,H,L,32]
// variant 1: gk = log_sigmoid(c)/16 -> [B,H,L,32] (k recomputed in scan)
// variant 2: v  -> [B,H,L,64]
// variant 3: g  -> silu(c + bg) kept as [B,L,D]
// ---------------------------------------------------------------------------
__global__ __launch_bounds__(256) void k_gemm_proj(const float* __restrict__ X,
                                                   const float* __restrict__ W,
                                                   const float* __restrict__ bias,
                                                   float* __restrict__ out0,
                                                   int N, int variant) {
  __shared__ __attribute__((aligned(16))) float Xs[64][68];
  __shared__ __attribute__((aligned(16))) float Ws[64][68];
  const int tid = threadIdx.x, lane = tid & 31, w = tid >> 5;
  const int m0 = blockIdx.x * 64, n0 = blockIdx.y * 64;
  const int mo = (w & 3) * 16;
  const int no0 = (w >> 2) * 32, no1 = no0 + 16;
  v8f acc0 = {}, acc1 = {};

  for (int k0 = 0; k0 < Dmod; k0 += 64) {
    for (int i = tid; i < 1024; i += 256) {
      int row = i >> 4, c4 = (i & 15) * 4;
      async_b128(&Xs[row][c4], X + (size_t)(m0 + row) * Dmod + k0 + c4);
      async_b128(&Ws[row][c4], W + (size_t)(k0 + row) * N + n0 + c4);
    }
    async_wait0();
    __syncthreads();
#pragma unroll
    for (int kk = 0; kk < 64; kk += 4) {
      v2f a  = ld_a(&Xs[0][0], 68, mo, kk, lane);
      v2f b0 = ld_b(&Ws[0][0], 68, kk, no0, lane);
      v2f b1 = ld_b(&Ws[0][0], 68, kk, no1, lane);
      acc0 = wmma4(a, b0, acc0);
      acc1 = wmma4(a, b1, acc1);
    }
    __syncthreads();
  }

  // epilogue
  const int nn[2] = {n0 + no0 + (lane & 15), n0 + no1 + (lane & 15)};
  const int mb = (lane >> 4) * 8;
#pragma unroll
  for (int half = 0; half < 2; half++) {
    v8f acc = half ? acc1 : acc0;
#pragma unroll
    for (int i = 0; i < 8; i++) {
      int r = m0 + mo + mb + i;
      int n = nn[half];
      float val = acc[i];
      int b = r >> 13, l = r & (Lseq - 1);
      if (variant == 0) {
        int h = n >> 5, j = n & 31;
        out0[(((size_t)(b * Hh + h)) * Lseq + l) * DK + j] = val;
      } else if (variant == 1) {
        int h = n >> 5, j = n & 31;
        out0[(((size_t)(b * Hh + h)) * Lseq + l) * DK + j] = logsigf_(val) * (1.0f / 16.0f);
      } else if (variant == 2) {
        int h = n >> 6, j = n & 63;
        out0[(((size_t)(b * Hh + h)) * Lseq + l) * DV + j] = val;
      } else {
        val += bias[n];
        out0[(size_t)r * Dmod + n] = siluf_(val);
      }
    }
  }
}

// ---------------------------------------------------------------------------
// K3: chunked GLA scan. One workgroup per (b,h); 128 sequential chunks of 64.
// Within a chunk (b = inclusive cumsum of gk):
//   P[t][s] = (q_t*scale*e^{b_t}) . (k_s*e^{-b_s})   masked s<=t
//   O = P @ V + (q*scale*e^{b}) @ S
//   S <- e^{b_C} * (S + Ke^T @ V)          (Ke = k*e^{-b})
// then o += sigmoid(augw*v), LayerNorm over dv, store to o[B,L,D] stripe.
// ---------------------------------------------------------------------------
__global__ __launch_bounds__(256) void k_gla_scan(const float* __restrict__ qg,
                                                  const float* __restrict__ gkg,
                                                  const float* __restrict__ vg,
                                                  const float* __restrict__ aug,
                                                  float* __restrict__ og,
                                                  float* __restrict__ stateg) {
  __shared__ __attribute__((aligned(16))) float Qm[64][36];   // q -> Qe
  __shared__ __attribute__((aligned(16))) float Km[64][36];   // raw gk -> Ke
  __shared__ __attribute__((aligned(16))) float Vm[64][64];
  __shared__ float Pm[64][66];                                // Bc -> P -> O
  __shared__ float Sm[32][66];                                // running state
  __shared__ float augw[64];
  __shared__ float bC[32];

  const int tid = threadIdx.x, lane = tid & 31, w = tid >> 5;
  const int bh = blockIdx.x, b = bh >> 4, h = bh & 15;
  const float* qb  = qg  + (size_t)bh * Lseq * DK;
  const float* gkb = gkg + (size_t)bh * Lseq * DK;
  const float* vb  = vg  + (size_t)bh * Lseq * DV;
  float* ob = og + (size_t)b * Lseq * Dmod + h * DV;
  const float* av = aug + h * DK;
  const float scale = 0.17677669529663687f;                   // 1/sqrt(32)

  for (int i = tid; i < 32 * 66; i += 256) (&Sm[0][0])[i] = 0.0f;
  __syncthreads();

  for (int c = 0; c < NCHUNK; c++) {
    const size_t o32 = (size_t)c * CHUNK * DK;
    const size_t o64 = (size_t)c * CHUNK * DV;

    // ---- async-stage chunk into LDS (q -> Qm, raw gk -> Km, v -> Vm)
    for (int i = tid; i < 512; i += 256) {
      int t = i >> 3, c4 = (i & 7) * 4;
      async_b128(&Qm[t][c4], qb + o32 + (size_t)t * DK + c4);
      async_b128(&Km[t][c4], gkb + o32 + (size_t)t * DK + c4);
    }
    for (int i = tid; i < 1024; i += 256) {
      int t = i >> 4, c4 = (i & 15) * 4;
      async_b128(&Vm[t][c4], vb + o64 + (size_t)t * DV + c4);
    }
    async_wait0();
    __syncthreads();

    // ---- expand: Pm = raw gk (cumsum workspace), Km = k = 1 - exp(gk)
    for (int i = tid; i < 64 * 32; i += 256) {
      int t = i >> 5, j = i & 31;
      float gv = Km[t][j];
      Pm[t][j] = gv;
      Km[t][j] = 1.0f - expf(gv);
    }
    __syncthreads();

    // ---- augw (raw q,k) and per-dim inclusive cumsum of gk
    if (tid < 64) {
      float s = 0.0f;
      for (int j = 0; j < DK; j++) s = fmaf(Qm[tid][j] * av[j], Km[tid][j], s);
      augw[tid] = s;
    } else if (tid < 96) {
      int j = tid - 64;
      float run = 0.0f;
      for (int t = 0; t < 64; t++) { run += Pm[t][j]; Pm[t][j] = run; }
      bC[j] = run;
    }
    __syncthreads();

    // ---- fold decay into q,k in place: Qe = q*scale*e^b, Ke = k*e^{-b}
    for (int i = tid; i < 64 * 32; i += 256) {
      int t = i >> 5, j = i & 31;
      float bv = Pm[t][j];
      Qm[t][j] *= scale * expf(bv);
      Km[t][j] *= expf(-bv);
    }
    __syncthreads();

    // ---- P = Qe @ Ke^T (64x64, K=32), causal-masked into Pm
    {
      int ti = w * 2;
      int mt = (ti >> 2) * 16, nt0 = (ti & 3) * 16, nt1 = nt0 + 16;
      v8f p0 = {}, p1 = {};
#pragma unroll
      for (int kk = 0; kk < 32; kk += 4) {
        v2f a  = ld_a (&Qm[0][0], 36, mt,  kk, lane);
        v2f b0 = ld_bt(&Km[0][0], 36, kk, nt0, lane);
        v2f b1 = ld_bt(&Km[0][0], 36, kk, nt1, lane);
        p0 = wmma4(a, b0, p0);
        p1 = wmma4(a, b1, p1);
      }
      int n = lane & 15, mb = (lane >> 4) * 8;
#pragma unroll
      for (int i = 0; i < 8; i++) {
        int t = mt + mb + i;
        Pm[t][nt0 + n] = (t >= nt0 + n) ? p0[i] : 0.0f;
        Pm[t][nt1 + n] = (t >= nt1 + n) ? p1[i] : 0.0f;
      }
    }
    __syncthreads();

    // ---- O = P @ V (K=64) + Qe @ S (K=32)
    v8f oa = {}, obv = {};
    const int mt = ((w * 2) >> 2) * 16, nt0 = ((w * 2) & 3) * 16, nt1 = nt0 + 16;
#pragma unroll
    for (int kk = 0; kk < 64; kk += 4) {
      v2f a  = ld_a(&Pm[0][0], 66, mt, kk, lane);
      v2f b0 = ld_b(&Vm[0][0], 64, kk, nt0, lane);
      v2f b1 = ld_b(&Vm[0][0], 64, kk, nt1, lane);
      oa  = wmma4(a, b0, oa);
      obv = wmma4(a, b1, obv);
    }
#pragma unroll
    for (int kk = 0; kk < 32; kk += 4) {
      v2f a  = ld_a(&Qm[0][0], 36, mt, kk, lane);
      v2f b0 = ld_b(&Sm[0][0], 66, kk, nt0, lane);
      v2f b1 = ld_b(&Sm[0][0], 66, kk, nt1, lane);
      oa  = wmma4(a, b0, oa);
      obv = wmma4(a, b1, obv);
    }
    __syncthreads();   // everyone done reading Pm (P) and Sm

    // ---- store O into Pm; update S tile (one 16x16 per wave)
    {
      int n = lane & 15, mb = (lane >> 4) * 8;
#pragma unroll
      for (int i = 0; i < 8; i++) {
        Pm[mt + mb + i][nt0 + n] = oa[i];
        Pm[mt + mb + i][nt1 + n] = obv[i];
      }
      int sm0 = (w & 1) * 16, sn0 = (w >> 1) * 16;
      v8f sacc = {};
#pragma unroll
      for (int kk = 0; kk < 64; kk += 4) {
        v2f a = ld_at(&Km[0][0], 36, sm0, kk, lane);   // A[j][t] = Ke[t][j]
        v2f bb = ld_b(&Vm[0][0], 64, kk, sn0, lane);
        sacc = wmma4(a, bb, sacc);
      }
#pragma unroll
      for (int i = 0; i < 8; i++) {
        int j = sm0 + mb + i, nn2 = sn0 + n;
        Sm[j][nn2] = expf(bC[j]) * (Sm[j][nn2] + sacc[i]);
      }
    }
    __syncthreads();

    // ---- epilogue: o += sigmoid(augw*v); LayerNorm over dv; store stripe
    if (tid < 64) {
      int t = tid;
      float aw = augw[t], sum = 0.0f, sq = 0.0f;
      for (int n2 = 0; n2 < DV; n2++) {
        float val = Pm[t][n2] + sigmoidf_(aw * Vm[t][n2]);
        Pm[t][n2] = val;
        sum += val; sq = fmaf(val, val, sq);
      }
      float mu = sum * (1.0f / DV);
      float var = sq * (1.0f / DV) - mu * mu;
      float rs = rsqrtf(var + 1e-5f);
      float* orow = ob + ((size_t)c * CHUNK + t) * Dmod;
      for (int n2 = 0; n2 < DV; n2++) orow[n2] = (Pm[t][n2] - mu) * rs;
    }
    __syncthreads();
  }

  // final state [B,H,32,64]
  for (int i = tid; i < DK * DV; i += 256) {
    stateg[(size_t)bh * (DK * DV) + i] = Sm[i >> 6][i & 63];
  }
}

// ---------------------------------------------------------------------------
// K4: out = (silu(g) * o) @ Wout; all three tiles async-staged; the
//     silu(g)*o product is fused into the A-fragment load; store transposed
//     to [L,B,D].
// ---------------------------------------------------------------------------
__global__ __launch_bounds__(256) void k_gemm_out(const float* __restrict__ sg,
                                                  const float* __restrict__ o,
                                                  const float* __restrict__ W,
                                                  float* __restrict__ out) {
  __shared__ __attribute__((aligned(16))) float Gs[64][68];
  __shared__ __attribute__((aligned(16))) float Os[64][68];
  __shared__ __attribute__((aligned(16))) float Ws[64][68];
  const int tid = threadIdx.x, lane = tid & 31, w = tid >> 5;
  const int m0 = blockIdx.x * 64, n0 = blockIdx.y * 64;
  const int mo = (w & 3) * 16;
  const int no0 = (w >> 2) * 32, no1 = no0 + 16;
  v8f acc0 = {}, acc1 = {};
  const int am = lane & 15, akh = lane >> 4;

  for (int k0 = 0; k0 < Dmod; k0 += 64) {
    for (int i = tid; i < 1024; i += 256) {
      int row = i >> 4, c4 = (i & 15) * 4;
      size_t gi = (size_t)(m0 + row) * Dmod + k0 + c4;
      async_b128(&Gs[row][c4], sg + gi);
      async_b128(&Os[row][c4], o + gi);
      async_b128(&Ws[row][c4], W + (size_t)(k0 + row) * Dmod + n0 + c4);
    }
    async_wait0();
    __syncthreads();
#pragma unroll
    for (int kk = 0; kk < 64; kk += 4) {
      int ai = (mo + am) * 68 + kk + akh * 2;
      v2f a;
      a.x = Gs[0][ai]     * Os[0][ai];        // silu(g)*o fused at frag load
      a.y = Gs[0][ai + 1] * Os[0][ai + 1];
      v2f b0 = ld_b(&Ws[0][0], 68, kk, no0, lane);
      v2f b1 = ld_b(&Ws[0][0], 68, kk, no1, lane);
      acc0 = wmma4(a, b0, acc0);
      acc1 = wmma4(a, b1, acc1);
    }
    __syncthreads();
  }

  const int mb = (lane >> 4) * 8;
#pragma unroll
  for (int half = 0; half < 2; half++) {
    v8f acc = half ? acc1 : acc0;
    int n = n0 + (half ? no1 : no0) + (lane & 15);
#pragma unroll
    for (int i = 0; i < 8; i++) {
      int r = m0 + mo + mb + i;
      int b = r >> 13, l = r & (Lseq - 1);
      out[((size_t)l * Bsz + b) * Dmod + n] = acc[i];
    }
  }
}

// ---------------------------------------------------------------------------
extern "C" void kernel_launch(void* const* d_in, const int* in_sizes, int n_in,
                              void* d_out, int out_size, void* d_ws, size_t ws_size,
                              hipStream_t stream) {
  const float* x      = (const float*)d_in[0];
  const float* conv_w = (const float*)d_in[1];
  const float* Wq     = (const float*)d_in[2];
  const float* Wkg    = (const float*)d_in[3];
  const float* Wv     = (const float*)d_in[4];
  const float* Wg     = (const float*)d_in[5];
  const float* bg     = (const float*)d_in[6];
  const float* Wout   = (const float*)d_in[7];
  const float* aug    = (const float*)d_in[8];

  float* out_seq   = (float*)d_out;                         // [L,B,D]
  float* out_state = out_seq + (size_t)Lseq * Bsz * Dmod;   // [B,H,32,64]

  // workspace partition (floats): xb/o share one region (xb dead after proj)
  float* xb  = (float*)d_ws;                           // 67,108,864
  float* qb  = xb  + (size_t)Bsz * Lseq * Dmod;        // 33,554,432
  float* gkb = qb  + (size_t)Bsz * Lseq * (Dmod / 2);  // 33,554,432
  float* vbf = gkb + (size_t)Bsz * Lseq * (Dmod / 2);  // 67,108,864
  float* sgb = vbf + (size_t)Bsz * Lseq * Dmod;        // 67,108,864
  float* obf = xb;                                     // reuse xb region

  k_conv_silu<<<(Bsz * Lseq * Dmod) / 256, 256, 0, stream>>>(x, conv_w, xb);

  k_gemm_proj<<<dim3(Mrows / 64, 512 / 64),  256, 0, stream>>>(xb, Wq,  nullptr, qb,  512,  0);
  k_gemm_proj<<<dim3(Mrows / 64, 512 / 64),  256, 0, stream>>>(xb, Wkg, nullptr, gkb, 512,  1);
  k_gemm_proj<<<dim3(Mrows / 64, 1024 / 64), 256, 0, stream>>>(xb, Wv,  nullptr, vbf, 1024, 2);
  k_gemm_proj<<<dim3(Mrows / 64, 1024 / 64), 256, 0, stream>>>(xb, Wg,  bg,      sgb, 1024, 3);

  k_gla_scan<<<Bsz * Hh, 256, 0, stream>>>(qb, gkb, vbf, aug, obf, out_state);

  k_gemm_out<<<dim3(Mrows / 64, 1024 / 64), 256, 0, stream>>>(sgb, obf, Wout, out_seq);
}